// RelativeDepthLoss_4724464025783
// MI455X (gfx1250) — compile-verified
//
#include <hip/hip_runtime.h>
#include <hip/hip_bf16.h>
#include <math.h>

// Problem constants from the reference (fixed by the harness).
#define B_  64
#define H_  512
#define W_  512
#define K_  10000
#define NBLK 512
#define NTHR 256

typedef __attribute__((ext_vector_type(2))) float v2f;
typedef __attribute__((ext_vector_type(8))) float v8f;

// Exact f32 wave32 reduction using V_WMMA_F32_16X16X4_F32 with an all-ones
// B matrix. A layout (16x4 f32): lanes 0-15 hold K=0 (VGPR0) / K=1 (VGPR1),
// lanes 16-31 hold K=2 / K=3. With a = {acc, 0}, row m of A sums to
// acc[m] + acc[m+16], replicated across all 16 columns of D. Summing the 8
// D VGPRs per lane gives rows 0-7 (lanes 0-15) or rows 8-15 (lanes 16-31);
// one xor-16 shuffle completes the full 32-lane sum (broadcast to all lanes).
// Requires EXEC == all 1s at the call site.
__device__ __forceinline__ float wave_reduce_f32_wmma(float acc) {
    v2f a; a[0] = acc;  a[1] = 0.0f;
    v2f b; b[0] = 1.0f; b[1] = 1.0f;
    v8f c = {};
    c = __builtin_amdgcn_wmma_f32_16x16x4_f32(
        /*neg_a=*/false, a, /*neg_b=*/false, b,
        /*c_mod=*/(short)0, c, /*reuse_a=*/false, /*reuse_b=*/false);
    float r = ((c[0] + c[1]) + (c[2] + c[3])) + ((c[4] + c[5]) + (c[6] + c[7]));
    r += __shfl_xor(r, 16, 32);
    return r;
}

__global__ void rdl_partial_kernel(const float* __restrict__ depth,
                                   const int*   __restrict__ xA,
                                   const int*   __restrict__ yA,
                                   const int*   __restrict__ xB,
                                   const int*   __restrict__ yB,
                                   const float* __restrict__ ord,
                                   float* __restrict__ partial,
                                   int n) {
    __shared__ float s_wave[NTHR / 32];

    float acc = 0.0f;
    const int stride = gridDim.x * blockDim.x;
    for (int i = blockIdx.x * blockDim.x + threadIdx.x; i < n; i += stride) {
        const int b = i / K_;                       // constant divisor -> mulhi
        const float* img = depth + (size_t)b * (H_ * W_);
        const float zA = img[xA[i] * W_ + yA[i]];   // random gather, L2-resident
        const float zB = img[xB[i] * W_ + yB[i]];
        const float t    = ord[i];
        const float mask = fabsf(t);
        const float d    = zA - zB;
        const float ll   = log1pf(expf(-d * t)) * mask;
        acc += ll + d * d * (1.0f - mask);
    }

    // All lanes converge here; EXEC is full for the WMMA reduction.
    const float wsum = wave_reduce_f32_wmma(acc);

    const int lane = threadIdx.x & 31;
    const int wid  = threadIdx.x >> 5;
    if (lane == 0) s_wave[wid] = wsum;
    __syncthreads();

    if (threadIdx.x == 0) {
        float s = 0.0f;
#pragma unroll
        for (int j = 0; j < NTHR / 32; ++j) s += s_wave[j];   // fixed order: deterministic
        partial[blockIdx.x] = s;
    }
}

// Single-wave final reduction: fixed lane-strided accumulation over the 512
// block partials, then the same exact WMMA wave reduction. Deterministic.
__global__ void rdl_final_kernel(const float* __restrict__ partial,
                                 float* __restrict__ out) {
    float acc = 0.0f;
    for (int j = threadIdx.x; j < NBLK; j += 32) acc += partial[j];
    const float s = wave_reduce_f32_wmma(acc);
    if (threadIdx.x == 0) out[0] = s / (float)B_;
}

extern "C" void kernel_launch(void* const* d_in, const int* in_sizes, int n_in,
                              void* d_out, int out_size, void* d_ws, size_t ws_size,
                              hipStream_t stream) {
    const float* depth = (const float*)d_in[0];   // (B,1,H,W) f32
    const int*   xA    = (const int*)  d_in[1];   // (B,K) i32
    const int*   yA    = (const int*)  d_in[2];
    const int*   xB    = (const int*)  d_in[3];
    const int*   yB    = (const int*)  d_in[4];
    const float* ord   = (const float*)d_in[5];   // (B,K) f32

    float* out     = (float*)d_out;
    float* partial = (float*)d_ws;                // NBLK floats = 2 KiB scratch
    const int n = in_sizes[1];                    // B*K = 640000

    rdl_partial_kernel<<<NBLK, NTHR, 0, stream>>>(depth, xA, yA, xB, yB, ord,
                                                  partial, n);
    rdl_final_kernel<<<1, 32, 0, stream>>>(partial, out);
}